// Model_50268297232523
// MI455X (gfx1250) — compile-verified
//
#include <hip/hip_runtime.h>

#define H  20
#define TT 512
#define BB 4096

typedef float v2f __attribute__((ext_vector_type(2)));
typedef float v4f __attribute__((ext_vector_type(4)));
typedef float v8f __attribute__((ext_vector_type(8)));

__device__ __forceinline__ v8f wmma4(v2f a, v2f b, v8f c) {
    // D = A(16x4 f32) x B(4x16 f32) + C(16x16 f32)
    return __builtin_amdgcn_wmma_f32_16x16x4_f32(
        /*neg_a=*/false, a, /*neg_b=*/false, b,
        /*c_mod=*/(short)0, c, /*reuse_a=*/false, /*reuse_b=*/false);
}

// gfx1250 native V_TANH_F32 (TRANS pipe); branch-free fallback otherwise.
__device__ __forceinline__ float fast_tanh(float x) {
#if __has_builtin(__builtin_amdgcn_tanhf)
    return __builtin_amdgcn_tanhf(x);
#else
    const float a = __builtin_fabsf(x);
    const float e = __builtin_amdgcn_exp2f(-2.885390081777927f * a);
    const float r = (1.0f - e) * __builtin_amdgcn_rcpf(1.0f + e);
    return __builtin_copysignf(r, x);
#endif
}

__global__ __launch_bounds__(32)
void rnn2_wmma_kernel(const float* __restrict__ x,      // [B,T]
                      const float* __restrict__ hidden, // [2,B,H]
                      const float* __restrict__ W_ih0,  // [H,1]
                      const float* __restrict__ W_hh0,  // [H,H]
                      const float* __restrict__ b_ih0,
                      const float* __restrict__ b_hh0,
                      const float* __restrict__ W_ih1,  // [H,H]
                      const float* __restrict__ W_hh1,  // [H,H]
                      const float* __restrict__ b_ih1,
                      const float* __restrict__ b_hh1,
                      const float* __restrict__ fc_w,   // [H]
                      const float* __restrict__ fc_b,   // [1]
                      float* __restrict__ out)          // [BB] ++ [2,BB,H]
{
    // Transposed state Hs[b][h], h contiguous. Row 16 is a write-only dummy
    // row so tile-1 stores need no EXEC branching (row stride 80B keeps all
    // rows 16B-aligned -> ds_store_b128 merge survives).
    __shared__ float Hs0[17][H];
    __shared__ float Hs1[17][H];

    const int lane = threadIdx.x & 31;
    const int n    = lane & 15;   // batch column (N index / A's M index)
    const int hi   = lane >> 4;   // lane half
    const int b0   = blockIdx.x * 16;
    const int rowT = (hi == 0) ? n : 16;   // tile-1 store row (dummy for hi=1)

    // ---- Loop-invariant weight A-fragments -------------------------------
    // A 16x4 f32 layout: lanes 0-15 hold M=lane, K={0,1}; lanes 16-31 K={2,3}
    v2f Ah0[2][5], Ai1[2][5], Ah1[2][5];
    #pragma unroll
    for (int tau = 0; tau < 2; ++tau) {
        const int mg = n + 16 * tau;       // output hidden index
        #pragma unroll
        for (int c = 0; c < 5; ++c) {
            const int k = 4 * c + 2 * hi;  // input hidden index (<= 19)
            v2f z; z.x = 0.f; z.y = 0.f;
            Ah0[tau][c] = z; Ai1[tau][c] = z; Ah1[tau][c] = z;
            if (mg < H) {
                Ah0[tau][c].x = W_hh0[mg * H + k];     Ah0[tau][c].y = W_hh0[mg * H + k + 1];
                Ai1[tau][c].x = W_ih1[mg * H + k];     Ai1[tau][c].y = W_ih1[mg * H + k + 1];
                Ah1[tau][c].x = W_hh1[mg * H + k];     Ah1[tau][c].y = W_hh1[mg * H + k + 1];
            }
        }
    }

    // Per-lane C-layout coefficients: element r <-> h = r + 8*hi + 16*tau
    v8f w0v[2], bi0v[2], bi1v[2];
    #pragma unroll
    for (int tau = 0; tau < 2; ++tau) {
        #pragma unroll
        for (int r = 0; r < 8; ++r) {
            const int h = r + 8 * hi + 16 * tau;
            const bool ok = (h < H);
            w0v[tau][r]  = ok ? W_ih0[h] : 0.f;
            bi0v[tau][r] = ok ? (b_ih0[h] + b_hh0[h]) : 0.f;
            bi1v[tau][r] = ok ? (b_ih1[h] + b_hh1[h]) : 0.f;
        }
    }
    v8f zero8;
    #pragma unroll
    for (int r = 0; r < 8; ++r) zero8[r] = 0.f;

    // ---- Initial hidden state into LDS -----------------------------------
    #pragma unroll
    for (int j = 0; j < 10; ++j) {
        const int h = hi * 10 + j;
        Hs0[n][h] = hidden[0 * BB * H + (b0 + n) * H + h];
        Hs1[n][h] = hidden[1 * BB * H + (b0 + n) * H + h];
    }
    __syncthreads();

    // ---- Sequential scan over T (4 timesteps per x-load) ------------------
    for (int tq = 0; tq < TT / 4; ++tq) {
        const v4f xq = *(const v4f*)&x[(b0 + n) * TT + 4 * tq];

        #pragma unroll
        for (int u = 0; u < 4; ++u) {
            const float xv = xq[u];

            // Old states as B fragments (B 4x16: lanes 0-15 K={0,1}, 16-31 K={2,3})
            v2f B0[5], B1o[5];
            #pragma unroll
            for (int c = 0; c < 5; ++c) {
                B0[c]  = *(const v2f*)&Hs0[n][4 * c + 2 * hi];
                B1o[c] = *(const v2f*)&Hs1[n][4 * c + 2 * hi];
            }

            // Layer-0 accumulators (input projection + bias as rank-1 init)
            v8f a0t0 = w0v[0] * xv + bi0v[0];
            v8f a0t1 = w0v[1] * xv + bi0v[1];
            // Layer-1 hh chain is independent of newh0 -> runs alongside layer 0.
            v8f h1t0 = wmma4(Ah1[0][0], B1o[0], bi1v[0]);
            v8f h1t1 = wmma4(Ah1[1][0], B1o[0], bi1v[1]);
            #pragma unroll
            for (int c = 0; c < 5; ++c) {
                a0t0 = wmma4(Ah0[0][c], B0[c], a0t0);
                a0t1 = wmma4(Ah0[1][c], B0[c], a0t1);
                if (c >= 1) {
                    h1t0 = wmma4(Ah1[0][c], B1o[c], h1t0);
                    h1t1 = wmma4(Ah1[1][c], B1o[c], h1t1);
                }
            }
            #pragma unroll
            for (int r = 0; r < 8; ++r) a0t0[r] = fast_tanh(a0t0[r]);
            #pragma unroll
            for (int r = 0; r < 4; ++r) a0t1[r] = fast_tanh(a0t1[r]);

            // Store newh0 (C layout: m = r + 8*hi (+16 tile1), n = lane&15).
            // Per-wave DS ops are in-order, so old-state reads above are safe.
            #pragma unroll
            for (int j = 0; j < 4; ++j) {
                v2f p; p.x = a0t0[2 * j]; p.y = a0t0[2 * j + 1];
                *(v2f*)&Hs0[n][8 * hi + 2 * j] = p;
            }
            {
                v2f p; p.x = a0t1[0]; p.y = a0t1[1];
                *(v2f*)&Hs0[rowT][16] = p;
                p.x = a0t1[2]; p.y = a0t1[3];
                *(v2f*)&Hs0[rowT][18] = p;
            }
            __syncthreads();   // cross-lane visibility of newh0

            // newh0 as B fragments
            v2f B1n[5];
            #pragma unroll
            for (int c = 0; c < 5; ++c)
                B1n[c] = *(const v2f*)&Hs0[n][4 * c + 2 * hi];

            // Layer-1 ih chain (only 5 WMMAs + 1 add on the critical path)
            v8f i1t0 = wmma4(Ai1[0][0], B1n[0], zero8);
            v8f i1t1 = wmma4(Ai1[1][0], B1n[0], zero8);
            #pragma unroll
            for (int c = 1; c < 5; ++c) {
                i1t0 = wmma4(Ai1[0][c], B1n[c], i1t0);
                i1t1 = wmma4(Ai1[1][c], B1n[c], i1t1);
            }
            v8f a1t0 = i1t0 + h1t0;
            v8f a1t1 = i1t1 + h1t1;
            #pragma unroll
            for (int r = 0; r < 8; ++r) a1t0[r] = fast_tanh(a1t0[r]);
            #pragma unroll
            for (int r = 0; r < 4; ++r) a1t1[r] = fast_tanh(a1t1[r]);

            #pragma unroll
            for (int j = 0; j < 4; ++j) {
                v2f p; p.x = a1t0[2 * j]; p.y = a1t0[2 * j + 1];
                *(v2f*)&Hs1[n][8 * hi + 2 * j] = p;
            }
            {
                v2f p; p.x = a1t1[0]; p.y = a1t1[1];
                *(v2f*)&Hs1[rowT][16] = p;
                p.x = a1t1[2]; p.y = a1t1[3];
                *(v2f*)&Hs1[rowT][18] = p;
            }
            __syncthreads();   // next iteration reads both states
        }
    }

    // ---- Outputs: out[B] then new_hidden[2,B,H] --------------------------
    float* nh = out + BB;
    #pragma unroll
    for (int j = 0; j < 10; ++j) {
        const int h = hi * 10 + j;
        nh[0 * BB * H + (b0 + n) * H + h] = Hs0[n][h];
        nh[1 * BB * H + (b0 + n) * H + h] = Hs1[n][h];
    }
    if (hi == 0) {
        float s = fc_b[0];
        #pragma unroll
        for (int h = 0; h < H; ++h) s += Hs1[n][h] * fc_w[h];
        out[b0 + n] = s;
    }
}

extern "C" void kernel_launch(void* const* d_in, const int* in_sizes, int n_in,
                              void* d_out, int out_size, void* d_ws, size_t ws_size,
                              hipStream_t stream) {
    (void)in_sizes; (void)n_in; (void)out_size; (void)d_ws; (void)ws_size;
    const float* x      = (const float*)d_in[0];
    const float* hidden = (const float*)d_in[1];
    const float* W_ih0  = (const float*)d_in[2];
    const float* W_hh0  = (const float*)d_in[3];
    const float* b_ih0  = (const float*)d_in[4];
    const float* b_hh0  = (const float*)d_in[5];
    const float* W_ih1  = (const float*)d_in[6];
    const float* W_hh1  = (const float*)d_in[7];
    const float* b_ih1  = (const float*)d_in[8];
    const float* b_hh1  = (const float*)d_in[9];
    const float* fc_w   = (const float*)d_in[10];
    const float* fc_b   = (const float*)d_in[11];

    dim3 grid(BB / 16);   // 256 waves, one 16-row batch tile each
    dim3 block(32);       // one wave32 per workgroup -> barriers are ~free
    rnn2_wmma_kernel<<<grid, block, 0, stream>>>(
        x, hidden, W_ih0, W_hh0, b_ih0, b_hh0,
        W_ih1, W_hh1, b_ih1, b_hh1, fc_w, fc_b, (float*)d_out);
}